// SpatialTemporalInteractiveGCN_39934605918860
// MI455X (gfx1250) — compile-verified
//
#include <hip/hip_runtime.h>

// SpatialTemporalInteractiveGCN fused kernel for gfx1250 (MI455X).
//
// Math collapse: the reference overwrites `adj` with a tiled identity, so the
// "graph aggregation" is just a 2-step temporal average:
//     s[b,t,n,:] = 0.5*(x[b,t-1,n,:] + x[b,t,n,:])   (x[b,-1]=0)
//     h1 = s@w1^T + b1 ; h2 = s@w2^T + b2
//     out = LayerNorm(relu(h1*h2) + h1 + x) * gamma + beta
//
// Bandwidth-bound (~0.8 GB traffic vs 12.6 GFLOP) -> keep fp32, use
// V_WMMA_F32_16X16X4_F32 for the 64x64 matmuls. One wave = 16 tokens x 64
// outputs; K=64 in 16 steps of 4; 128 WMMA per tile.

typedef float v2f __attribute__((ext_vector_type(2)));
typedef float v8f __attribute__((ext_vector_type(8)));

#define DCH 64
#define NNODES 1000
#define TSTEPS 24
#define LN_EPS 1e-5f

__global__ __launch_bounds__(256, 2)
void stgcn_fused_wmma(const float* __restrict__ x,
                      const float* __restrict__ w1, const float* __restrict__ b1,
                      const float* __restrict__ w2, const float* __restrict__ b2,
                      const float* __restrict__ gamma, const float* __restrict__ beta,
                      float* __restrict__ out)
{
    const int lane = threadIdx.x & 31;
    const int wave = threadIdx.x >> 5;
    const int half = lane >> 4;   // 0: lanes 0-15, 1: lanes 16-31
    const int l16  = lane & 15;

    // Each wave processes 16 consecutive tokens; 8 waves/block -> 128 tokens.
    const int tileBase = blockIdx.x * 128 + wave * 16;

    // ---- A-operand (s matrix) addressing -----------------------------------
    // ISA A layout (16x4 f32): lane holds row M=l16; VGPR0 = K in {0,2},
    // VGPR1 = K in {1,3}, with the +2 selected by lane half.
    const int tokA = tileBase + l16;
    const int tA = (tokA / NNODES) % TSTEPS;
    const float pscale = (tA > 0) ? 0.5f : 0.0f;            // zero pad at t==0
    const float* curp = x + (size_t)tokA * DCH + 2 * half;
    const float* prvp = (tA > 0) ? (curp - (size_t)NNODES * DCH) : curp; // safe addr

    // ---- B-operand (weights, row-major [64][64]) ---------------------------
    // h[m][n] = sum_k s[m][k] * w[n][k]  =>  B[k][n] = w[n][k].
    // Mirrored striping: lane holds column N=l16(+16*nt); VGPR0/1 = K pair,
    // +2 selected by lane half.
    const float* w1p = w1 + (size_t)l16 * DCH + 2 * half;
    const float* w2p = w2 + (size_t)l16 * DCH + 2 * half;

    const v8f vzero = {0.f, 0.f, 0.f, 0.f, 0.f, 0.f, 0.f, 0.f};
    v8f acc1[4], acc2[4];
    #pragma unroll
    for (int nt = 0; nt < 4; ++nt) { acc1[nt] = vzero; acc2[nt] = vzero; }

    // ---- K loop: 16 steps of K=4, 8 WMMAs per step -------------------------
    #pragma unroll
    for (int kk = 0; kk < 16; ++kk) {
        const int koff = kk * 4;
        v2f c = *(const v2f*)(curp + koff);
        v2f p = *(const v2f*)(prvp + koff);
        v2f a;
        a.x = 0.5f * c.x + pscale * p.x;
        a.y = 0.5f * c.y + pscale * p.y;
        #pragma unroll
        for (int nt = 0; nt < 4; ++nt) {
            v2f bw1 = *(const v2f*)(w1p + (size_t)nt * 16 * DCH + koff);
            v2f bw2 = *(const v2f*)(w2p + (size_t)nt * 16 * DCH + koff);
            acc1[nt] = __builtin_amdgcn_wmma_f32_16x16x4_f32(
                false, a, false, bw1, (short)0, acc1[nt], false, false);
            acc2[nt] = __builtin_amdgcn_wmma_f32_16x16x4_f32(
                false, a, false, bw2, (short)0, acc2[nt], false, false);
        }
    }

    // ---- Per-column constants (n = l16 + 16*nt) ----------------------------
    float bb1[4], bb2[4], gg[4], bt[4];
    #pragma unroll
    for (int nt = 0; nt < 4; ++nt) {
        const int n = l16 + 16 * nt;
        bb1[nt] = b1[n]; bb2[nt] = b2[n]; gg[nt] = gamma[n]; bt[nt] = beta[n];
    }

    // ---- Epilogue: gate + residual + LayerNorm + store ---------------------
    // C/D layout: lane half selects M in [0,8)+8*half via VGPR index r;
    // each lane holds 4 columns (one per nt). Row stats need a 16-lane
    // butterfly (masks 1,2,4,8 stay inside the half-wave on wave32).
    #pragma unroll
    for (int r = 0; r < 8; ++r) {
        const int m = r + 8 * half;
        const size_t rowOff = (size_t)(tileBase + m) * DCH;
        float o[4];
        float sum = 0.f, sq = 0.f;
        #pragma unroll
        for (int nt = 0; nt < 4; ++nt) {
            const float h1v = acc1[nt][r] + bb1[nt];
            const float h2v = acc2[nt][r] + bb2[nt];
            float gte = h1v * h2v;
            gte = gte > 0.f ? gte : 0.f;
            const float ov = gte + h1v + x[rowOff + l16 + 16 * nt];
            o[nt] = ov;
            sum += ov;
            sq  += ov * ov;
        }
        #pragma unroll
        for (int mk = 1; mk < 16; mk <<= 1) {
            sum += __shfl_xor(sum, mk, 32);
            sq  += __shfl_xor(sq,  mk, 32);
        }
        const float mu  = sum * (1.0f / DCH);
        const float var = sq * (1.0f / DCH) - mu * mu;
        const float inv = rsqrtf(var + LN_EPS);
        #pragma unroll
        for (int nt = 0; nt < 4; ++nt) {
            out[rowOff + l16 + 16 * nt] = (o[nt] - mu) * inv * gg[nt] + bt[nt];
        }
    }
}

extern "C" void kernel_launch(void* const* d_in, const int* in_sizes, int n_in,
                              void* d_out, int out_size, void* d_ws, size_t ws_size,
                              hipStream_t stream) {
    const float* x     = (const float*)d_in[0];
    // d_in[1] = adj : DEAD in the reference (overwritten with tiled identity).
    const float* w1    = (const float*)d_in[2];
    const float* b1    = (const float*)d_in[3];
    const float* w2    = (const float*)d_in[4];
    const float* b2    = (const float*)d_in[5];
    const float* gamma = (const float*)d_in[6];
    const float* beta  = (const float*)d_in[7];
    float* out = (float*)d_out;

    const int total_tokens = in_sizes[0] / DCH;        // B*T*N = 768000
    const int blocks = total_tokens / 128;             // 6000, exact
    stgcn_fused_wmma<<<blocks, 256, 0, stream>>>(x, w1, b1, w2, b2, gamma, beta, out);
}